// LiquidStateMachine_30580167147916
// MI455X (gfx1250) — compile-verified
//
#include <hip/hip_runtime.h>
#include <math.h>

// Liquid State Machine on MI455X (gfx1250, wave32).
// Recurrent GEMM via V_WMMA_F32_16X16X4_F32 (full f32 precision — workload is
// latency bound on the sequential T=200 recurrence; everything is L2-resident,
// so we optimize the per-step critical path: the K=2256 reduction is split
// across 4 waves per 32x16 output tile and combined through LDS).

typedef __attribute__((ext_vector_type(2))) float v2f;
typedef __attribute__((ext_vector_type(8))) float v8f;

#define LSM_B  32
#define LSM_T  200
#define LSM_NI 256
#define LSM_R  2000
#define LSM_NO 10

__global__ void lsm_zero_kernel(float* __restrict__ p, int n) {
  int i = blockIdx.x * blockDim.x + threadIdx.x;
  if (i < n) p[i] = 0.0f;
}

__device__ __forceinline__ void lsm_update(int idx, float I, float dwt,
    float* __restrict__ v, float* __restrict__ a, float* __restrict__ s_cur,
    float* __restrict__ sumv, float* __restrict__ sums, float* __restrict__ sumwv) {
  float vv = v[idx];
  float aa = a[idx];
  vv += (I - vv) * (1.0f / 20.0f);                 // v += (I - v)/TAU_MEM
  float spike = (vv >= 1.0f + aa) ? 1.0f : 0.0f;   // threshold w/ adaptation
  vv *= (1.0f - spike);                            // reset
  aa = aa * (1.0f - 1.0f / 100.0f) + 0.1f * spike; // adaptation decay + jump
  v[idx] = vv;
  a[idx] = aa;
  s_cur[idx] = spike;
  sumv[idx]  += vv;
  sums[idx]  += spike;
  sumwv[idx] += dwt * vv;
}

// 4 waves per block, one block per 16-neuron tile.  Each wave accumulates a
// K-slice of I = x_t@Win^T + s_prev@Wres^T for the full 32x16 tile with f32
// WMMA; partials are reduced through LDS and all 128 threads run the LIF
// epilogue cooperatively.  The wave id is forced into an SGPR via
// readfirstlane so all wave-level control flow is scalar branches — EXEC
// stays all-ones around every WMMA, as the ISA requires.
__global__ void __launch_bounds__(128) lsm_step_kernel(
    const float* __restrict__ x,      // (B, T, NI)
    const float* __restrict__ Win,    // (R, NI)
    const float* __restrict__ Wres,   // (R, R)
    const float* __restrict__ s_prev, // (B, R)
    float* __restrict__ s_cur,        // (B, R)
    float* __restrict__ v, float* __restrict__ a,
    float* __restrict__ sumv, float* __restrict__ sums, float* __restrict__ sumwv,
    int t, float dwt) {
  __shared__ float part[4][512];      // 4 waves x (32 lanes x 16 accum values)

  const int tid  = threadIdx.x;
  // wave id as a scalar (uniform) value -> scalar branches, no EXEC masking
  const int w    = __builtin_amdgcn_readfirstlane(tid >> 5);   // 0..3
  const int lane = tid & 31;          // all 32 lanes active around WMMA
  const int n0 = blockIdx.x * 16;     // neuron tile
  const int nn = lane & 15;
  const int h  = lane >> 4;

  v8f c0 = {};   // rows (batches) 0..15
  v8f c1 = {};   // rows (batches) 16..31

  // ---- Phase 1 (wave 0 only): I += x_t @ Win^T  (K = NI = 256) ----
  if (w == 0) {
    const size_t lda = (size_t)LSM_T * LSM_NI;
    // B fragment: lane holds Win[n0+nn, k+2h .. k+2h+1] (contiguous row of Win)
    const float* brow  = Win + (size_t)(n0 + nn) * LSM_NI + 2 * h;
    // A fragment: lane holds x[b=nn, t, k+2h .. k+2h+1]
    const float* arow0 = x + (size_t)nn * lda + (size_t)t * LSM_NI + 2 * h;
    const float* arow1 = arow0 + 16 * lda;
#pragma unroll 8
    for (int k = 0; k < LSM_NI; k += 4) {
      v2f bf = *(const v2f*)(brow  + k);
      v2f a0 = *(const v2f*)(arow0 + k);
      v2f a1 = *(const v2f*)(arow1 + k);
      c0 = __builtin_amdgcn_wmma_f32_16x16x4_f32(false, a0, false, bf, (short)0, c0, false, false);
      c1 = __builtin_amdgcn_wmma_f32_16x16x4_f32(false, a1, false, bf, (short)0, c1, false, false);
    }
  }

  // ---- Phase 2: I += s_prev @ Wres^T, K-slice per wave ----
  // K_total = 2256 split evenly: wave0 gets input(256) + reservoir [0,308);
  // waves 1..3 get reservoir slices of 564 elements each (all multiples of 4).
  {
    const int kr0 = (w == 0) ? 0   : 308 + (w - 1) * 564;   // scalar (w uniform)
    const int kr1 = (w == 0) ? 308 : kr0 + 564;
    const float* brow  = Wres   + (size_t)(n0 + nn) * LSM_R + 2 * h;
    const float* arow0 = s_prev + (size_t)nn * (size_t)LSM_R + 2 * h;
    const float* arow1 = arow0 + 16 * (size_t)LSM_R;
#pragma unroll 4
    for (int k = kr0; k < kr1; k += 4) {
      v2f bf = *(const v2f*)(brow  + k);
      v2f a0 = *(const v2f*)(arow0 + k);
      v2f a1 = *(const v2f*)(arow1 + k);
      c0 = __builtin_amdgcn_wmma_f32_16x16x4_f32(false, a0, false, bf, (short)0, c0, false, false);
      c1 = __builtin_amdgcn_wmma_f32_16x16x4_f32(false, a1, false, bf, (short)0, c1, false, false);
    }
  }

  // ---- Cross-wave reduction through LDS ----
  // Slot layout: part[w][lane*16 + j], j=0..7 -> c0[j], j=8..15 -> c1[j-8].
  {
    float* dst = &part[w][lane * 16];
#pragma unroll
    for (int j = 0; j < 8; ++j) {
      dst[j]     = c0[j];
      dst[8 + j] = c1[j];
    }
  }
  __syncthreads();

  // ---- Epilogue: each of the 128 threads reduces 4 slots + LIF update ----
  // Slot (origLane, j): c0[j] -> (row = (origLane>>4)*8 + j, col = origLane&15)
  //                     c1[j] -> row + 16.
  {
    const int origLane = tid >> 2;          // 0..31
    const int jbase    = (tid & 3) * 4;     // 0,4,8,12
    const int col = n0 + (origLane & 15);
    const int hh  = origLane >> 4;
#pragma unroll
    for (int jj = 0; jj < 4; ++jj) {
      const int j = jbase + jj;             // 0..15
      const int slot = origLane * 16 + j;
      float I = part[0][slot] + part[1][slot] + part[2][slot] + part[3][slot];
      const int jm  = j & 7;
      const int row = hh * 8 + jm + ((j >> 3) ? 16 : 0);
      lsm_update(row * LSM_R + col, I, dwt, v, a, s_cur, sumv, sums, sumwv);
    }
  }
}

// out[b,o] = sum_r 0.4*v_fin*W[o,r] + 0.3*(sumv/T)*W[o,R+r]
//          + 0.2*(sums/T)*W[o,2R+r] + 0.1*sumwv*W[o,3R+r] + bias[o]
__global__ void __launch_bounds__(32) lsm_readout_kernel(
    const float* __restrict__ v, const float* __restrict__ sumv,
    const float* __restrict__ sums, const float* __restrict__ sumwv,
    const float* __restrict__ Wro, const float* __restrict__ bro,
    float* __restrict__ out) {
  const int b = threadIdx.x;   // 32 batches, one per lane
  const int o = blockIdx.x;    // 10 outputs
  const float* w = Wro + (size_t)o * (4 * LSM_R);
  const size_t base = (size_t)b * LSM_R;
  const float invT = 1.0f / (float)LSM_T;
  float acc = bro[o];
  for (int r = 0; r < LSM_R; ++r) {
    acc += 0.4f *  v[base + r]            * w[r];
    acc += 0.3f * (sumv[base + r] * invT) * w[LSM_R + r];
    acc += 0.2f * (sums[base + r] * invT) * w[2 * LSM_R + r];
    acc += 0.1f *  sumwv[base + r]        * w[3 * LSM_R + r];
  }
  out[b * LSM_NO + o] = acc;
}

extern "C" void kernel_launch(void* const* d_in, const int* in_sizes, int n_in,
                              void* d_out, int out_size, void* d_ws, size_t ws_size,
                              hipStream_t stream) {
  const float* x    = (const float*)d_in[0]; // (B,T,NI)
  const float* Win  = (const float*)d_in[1]; // (R,NI)
  const float* Wres = (const float*)d_in[2]; // (R,R)
  const float* Wro  = (const float*)d_in[3]; // (NO,4R)
  const float* bro  = (const float*)d_in[4]; // (NO)
  float* out = (float*)d_out;
  float* ws  = (float*)d_ws;

  const size_t S = (size_t)LSM_B * LSM_R;    // 64000 floats per state array
  float* v     = ws;
  float* a     = ws + S;
  float* s0    = ws + 2 * S;
  float* s1    = ws + 3 * S;
  float* sumv  = ws + 4 * S;
  float* sums  = ws + 5 * S;
  float* sumwv = ws + 6 * S;

  {
    int n = (int)(7 * S);
    lsm_zero_kernel<<<(n + 255) / 256, 256, 0, stream>>>(ws, n);
  }

  // normalized exponential decay weights dw[t] = exp(-t/10)/Z
  double Z = 0.0;
  for (int t = 0; t < LSM_T; ++t) Z += exp(-(double)t / 10.0);

  for (int t = 0; t < LSM_T; ++t) {
    const float dwt = (float)(exp(-(double)t / 10.0) / Z);
    const float* sp = (t & 1) ? s1 : s0;   // spikes from previous step
    float*       sc = (t & 1) ? s0 : s1;   // spikes written this step
    lsm_step_kernel<<<LSM_R / 16, 128, 0, stream>>>(
        x, Win, Wres, sp, sc, v, a, sumv, sums, sumwv, t, dwt);
  }

  lsm_readout_kernel<<<LSM_NO, LSM_B, 0, stream>>>(v, sumv, sums, sumwv, Wro, bro, out);
}